// ReassignedSpectrogram_8040178778301
// MI455X (gfx1250) — compile-verified
//
#include <hip/hip_runtime.h>
#include <math.h>

// ---------------------------------------------------------------------------
// Reassigned spectrogram for MI455X (gfx1250, wave32, WMMA).
// DFT-as-GEMM through V_WMMA_F32_16X16X4_F32 (full fp32 matrix pipe).
// ---------------------------------------------------------------------------

typedef __attribute__((ext_vector_type(2))) float v2f;
typedef __attribute__((ext_vector_type(8))) float v8f;

#define N_FFT    1024
#define HOP      256
#define SIG_LEN  4194305
#define T_FRAMES 16385          // 1 + (SIG_LEN + N_FFT - N_FFT)/HOP
#define F_BINS   513            // N_FFT/2 + 1
#define F_TILES  33             // ceil(513/16)
#define T_TILES  1025           // ceil(16385/16)
#define SEG      4864           // 15*HOP + N_FFT samples per 16-frame tile
#define TWO_PI_F     6.28318530717958647692f
#define INV_TWO_PI_F 0.15915494309189533577f

// reflect-pad indexing: x_pad[j] = signal[reflect(j - 512)], single reflection
__device__ __forceinline__ int reflect_idx(int p) {
    if (p < 0) return -p;
    if (p >= SIG_LEN) return 2 * SIG_LEN - 2 - p;
    return p;
}

// ---------------------------------------------------------------------------
// Phase 1: windowed-frame DFT via fp32 WMMA.
// Block = 128 threads = 4 waves; block owns 16 frames, each wave a 16-freq tile.
// Produces spec_re, spec_im, and fused instantaneous frequency
//   instF = arg(spec * conj(spec_shift)) / 2pi  mod 1
// Planes stored [f * T_FRAMES + t].
// ---------------------------------------------------------------------------
__global__ __launch_bounds__(128)
void stft_wmma_kernel(const float* __restrict__ sig,
                      const float* __restrict__ window,
                      float* __restrict__ specRe,
                      float* __restrict__ specIm,
                      float* __restrict__ instF) {
    __shared__ float lsegN[SEG];   // reflected-padded signal segment
    __shared__ float lsegS[SEG];   // same, for 1-sample-delayed signal
    __shared__ float lwin[N_FFT];

    const int tid = threadIdx.x;
    const int t0  = blockIdx.x * 16;
    const int base = t0 * HOP - (N_FFT / 2);   // signal-domain index of seg[0]

    for (int i = tid; i < SEG; i += 128) {
        int p = reflect_idx(base + i);
        lsegN[i] = sig[p];
        lsegS[i] = (p == 0) ? 0.0f : sig[p - 1];   // shifted signal, then reflect
    }
    for (int i = tid; i < N_FFT; i += 128) lwin[i] = window[i];
    __syncthreads();

    const int wave = tid >> 5;
    const int lane = tid & 31;
    const int ft   = blockIdx.y * 4 + wave;       // wave-uniform
    if (ft >= F_TILES) return;                    // uniform branch: EXEC stays all-1
    const int f0 = ft * 16;

    // WMMA f32 16x16x4 fragment coords (wave32):
    //  A (16x4): lane m = lane&15; K pair = (lane>>4)*2 .. +1
    //  B (4x16): lane col = lane&15; rows = same K pair
    const int m    = lane & 15;
    const int kh   = (lane >> 4) << 1;            // 0 or 2
    const int fcol = f0 + (lane & 15);            // frequency for B column

    const float* segRowN = &lsegN[m * HOP];
    const float* segRowS = &lsegS[m * HOP];

    v8f accNre = {}; v8f accNim = {};
    v8f accSre = {}; v8f accSim = {};

    #pragma unroll 4
    for (int k0 = 0; k0 < N_FFT; k0 += 4) {
        const int k = k0 + kh;
        const float w0 = lwin[k], w1 = lwin[k + 1];
        v2f aN = { segRowN[k] * w0, segRowN[k + 1] * w1 };
        v2f aS = { segRowS[k] * w0, segRowS[k + 1] * w1 };

        // exact modular reduction of f*k before sincos -> fp32-accurate twiddles
        const int r0 = (fcol * k) & (N_FFT - 1);
        const int r1 = (fcol * (k + 1)) & (N_FFT - 1);
        float s0, c0, s1, c1;
        __sincosf((float)r0 * (TWO_PI_F / (float)N_FFT), &s0, &c0);
        __sincosf((float)r1 * (TWO_PI_F / (float)N_FFT), &s1, &c1);
        v2f bC = {  c0,  c1 };    // Re: cos
        v2f bS = { -s0, -s1 };    // Im: -sin  (rfft convention)

        accNre = __builtin_amdgcn_wmma_f32_16x16x4_f32(false, aN, false, bC,
                                                       (short)0, accNre, false, false);
        accNim = __builtin_amdgcn_wmma_f32_16x16x4_f32(false, aN, false, bS,
                                                       (short)0, accNim, false, false);
        accSre = __builtin_amdgcn_wmma_f32_16x16x4_f32(false, aS, false, bC,
                                                       (short)0, accSre, false, false);
        accSim = __builtin_amdgcn_wmma_f32_16x16x4_f32(false, aS, false, bS,
                                                       (short)0, accSim, false, false);
    }

    // C/D layout: VGPR r -> M = r (lanes 0-15) or r+8 (lanes 16-31); N = lane&15
    const int mhi  = (lane >> 4) * 8;
    const int fout = f0 + (lane & 15);
    if (fout < F_BINS) {
        #pragma unroll
        for (int r = 0; r < 8; ++r) {
            const int t = t0 + r + mhi;
            if (t < T_FRAMES) {
                const float nre = accNre[r], nim = accNim[r];
                const float sre = accSre[r], sim = accSim[r];
                // spec * conj(spec_shift)
                const float cre = nre * sre + nim * sim;
                const float cim = nim * sre - nre * sim;
                float a = atan2f(cim, cre) * INV_TWO_PI_F;
                a -= floorf(a);                    // mod 1 -> [0,1)
                const size_t idx = (size_t)fout * T_FRAMES + t;
                specRe[idx] = nre;
                specIm[idx] = nim;
                instF[idx]  = a;
            }
        }
    }
}

// ---------------------------------------------------------------------------
// Phase 2: reassignment + weighted 2D histogram (atomic fp32 adds).
// ---------------------------------------------------------------------------
__global__ __launch_bounds__(256)
void reassign_hist_kernel(const float* __restrict__ specRe,
                          const float* __restrict__ specIm,
                          const float* __restrict__ instF,
                          float* __restrict__ hist) {
    const int t = blockIdx.x * 256 + threadIdx.x;
    const int f = blockIdx.y;
    if (t >= T_FRAMES) return;

    const size_t idx = (size_t)f * T_FRAMES + t;
    const float re = specRe[idx], im = specIm[idx];
    const float mag = sqrtf(re * re + im * im) * (1.0f / (float)F_BINS);
    const float rf = instF[idx];

    float fre = 0.0f, fim = 0.0f;
    if (f > 0) { fre = specRe[idx - T_FRAMES]; fim = specIm[idx - T_FRAMES]; }
    // spec(f) * conj(spec(f-1))
    const float cre = re * fre + im * fim;
    const float cim = im * fre - re * fim;
    float a = atan2f(cim, cre) * INV_TWO_PI_F;
    a -= floorf(a);
    const float td = 0.5f - a;

    const float win_dur = (float)N_FFT / 44100.0f;
    const float eps     = 1.1920928955078125e-7f;   // FLT_EPSILON
    const float center  = (float)t * ((float)HOP / 44100.0f) + win_dur * 0.5f + eps;
    const float rt      = center + td * win_dur;

    const float f_hi = 0.5f;
    const float t_hi = (float)T_FRAMES * (float)HOP / 44100.0f;
    const float fi = rf / f_hi * (float)F_BINS;
    const float ti = rt / t_hi * (float)T_FRAMES;
    int bf = (rf == f_hi) ? (F_BINS - 1)   : (int)floorf(fi);
    int bt = (rt == t_hi) ? (T_FRAMES - 1) : (int)floorf(ti);
    const bool in_range = (rf >= 0.0f) && (rf <= f_hi) && (rt >= 0.0f) && (rt <= t_hi);
    if (in_range) {
        bf = min(max(bf, 0), F_BINS - 1);
        bt = min(max(bt, 0), T_FRAMES - 1);
        atomicAdd(&hist[(size_t)bf * T_FRAMES + bt], mag);
    }
}

// ---------------------------------------------------------------------------
// Utility kernels
// ---------------------------------------------------------------------------
__global__ __launch_bounds__(256)
void zero_kernel(float* __restrict__ p, size_t n) {
    const size_t i = (size_t)blockIdx.x * 256 + threadIdx.x;
    if (i < n) p[i] = 0.0f;
}

__global__ __launch_bounds__(256)
void db_kernel(const float* __restrict__ hist, float* __restrict__ out, size_t n) {
    const size_t i = (size_t)blockIdx.x * 256 + threadIdx.x;
    if (i < n) out[i] = 20.0f * log10f(fmaxf(1e-6f, hist[i]));
}

// ---------------------------------------------------------------------------
// Launch
// ---------------------------------------------------------------------------
extern "C" void kernel_launch(void* const* d_in, const int* in_sizes, int n_in,
                              void* d_out, int out_size, void* d_ws, size_t ws_size,
                              hipStream_t stream) {
    (void)in_sizes; (void)n_in; (void)out_size; (void)ws_size;
    const float* sig = (const float*)d_in[0];
    const float* win = (const float*)d_in[1];
    float* out = (float*)d_out;

    const size_t P = (size_t)F_BINS * (size_t)T_FRAMES;   // 8,405,505 elems / plane
    float* specRe = (float*)d_ws;
    float* specIm = specRe + P;
    float* instF  = specIm + P;
    float* hist   = instF + P;                            // 4 planes ~134.5 MB

    zero_kernel<<<dim3((unsigned)((P + 255) / 256)), dim3(256), 0, stream>>>(hist, P);

    dim3 g1(T_TILES, 9);   // 9 * 4 waves covers 33 freq tiles (guarded)
    stft_wmma_kernel<<<g1, dim3(128), 0, stream>>>(sig, win, specRe, specIm, instF);

    dim3 g2((T_FRAMES + 255) / 256, F_BINS);
    reassign_hist_kernel<<<g2, dim3(256), 0, stream>>>(specRe, specIm, instF, hist);

    db_kernel<<<dim3((unsigned)((P + 255) / 256)), dim3(256), 0, stream>>>(hist, out, P);
}